// Model_3813930959137
// MI455X (gfx1250) — compile-verified
//
#include <hip/hip_runtime.h>
#include <math.h>

#define F_ 16
#define L_ 8
#define V_ 8192
#define RS_ 512
#define H_ 720
#define W_ 1280
#define ORIG_ 1280
// Loss reads grid rows iy = (2r+1)/5, r<720  =>  iy <= 287. Rows >= 288 of
// every plane are never read, so they are neither zeroed nor link-summed.
#define ROWS_USED_ 288
#define PLANE_ (RS_ * RS_)              // allocation stride per (f,l) plane
#define USED_ (ROWS_USED_ * RS_)        // 147456 cells actually consumed

typedef __attribute__((ext_vector_type(2))) float v2f;
typedef __attribute__((ext_vector_type(8))) float v8f;

// ---------------------------------------------------------------------------
// Kernel A: se3_exp_map(dof) -> pose_bc; pose_ec[f,l] = pose_bc @ pose_ebs[f,l]
// 128 (f,l) pairs, one thread each. Thread 0 zeroes the loss accumulator.
// ---------------------------------------------------------------------------
__global__ void k_pose(const float* __restrict__ dof,
                       const float* __restrict__ pose_ebs,
                       float* __restrict__ pose_ec,
                       float* __restrict__ loss_acc)
{
    int t = threadIdx.x;
    if (t == 0) loss_acc[0] = 0.0f;
    if (t >= F_ * L_) return;

    float vx = dof[0], vy = dof[1], vz = dof[2];
    float wx = dof[3], wy = dof[4], wz = dof[5];
    float th2 = wx*wx + wy*wy + wz*wz;
    float th  = sqrtf(th2 + 1e-12f);
    float sn = sinf(th), cs = cosf(th);
    float Ac = sn / th;
    float Bc = (1.0f - cs) / (th * th);
    float Cc = (th - sn) / (th * th * th);

    float Wm[3][3] = {{0.f, -wz,  wy},
                      { wz, 0.f, -wx},
                      {-wy,  wx, 0.f}};
    float W2[3][3];
    for (int i = 0; i < 3; i++)
        for (int j = 0; j < 3; j++) {
            float s = 0.f;
            for (int k = 0; k < 3; k++) s += Wm[i][k] * Wm[k][j];
            W2[i][j] = s;
        }

    float bc[4][4];
    float tv[3];
    for (int i = 0; i < 3; i++) {
        float s = 0.f;
        for (int j = 0; j < 3; j++) {
            float I = (i == j) ? 1.f : 0.f;
            bc[i][j]  = I + Ac * Wm[i][j] + Bc * W2[i][j];
            float Vij = I + Bc * Wm[i][j] + Cc * W2[i][j];
            s += Vij * ((j == 0) ? vx : ((j == 1) ? vy : vz));
        }
        tv[i] = s;
    }
    bc[0][3] = tv[0]; bc[1][3] = tv[1]; bc[2][3] = tv[2];
    bc[3][0] = 0.f; bc[3][1] = 0.f; bc[3][2] = 0.f; bc[3][3] = 1.f;

    const float* eb = pose_ebs + t * 16;
    float* P = pose_ec + t * 12;
    for (int i = 0; i < 3; i++)
        for (int j = 0; j < 4; j++) {
            float s = 0.f;
            for (int k = 0; k < 4; k++) s += bc[i][k] * eb[k * 4 + j];
            P[i * 4 + j] = s;
        }
}

// ---------------------------------------------------------------------------
// Kernel B: zero only the first ROWS_USED_ rows of each of the 128 planes
// (75.5 MB instead of 134 MB) with wide float4 stores.
// ---------------------------------------------------------------------------
__global__ void k_zero(float* __restrict__ acc)
{
    const int perPlane4 = USED_ / 4;           // float4s per plane region
    const int n4 = F_ * L_ * perPlane4;
    int stride = gridDim.x * blockDim.x;
    for (int i = blockIdx.x * blockDim.x + threadIdx.x; i < n4; i += stride) {
        int plane = i / perPlane4;
        int off4  = i - plane * perPlane4;
        float4* p = (float4*)(acc + (long)plane * PLANE_) + off4;
        *p = make_float4(0.f, 0.f, 0.f, 0.f);
    }
}

// ---------------------------------------------------------------------------
// Kernel C: transform+project 1M points, bilinear splat (4 float atomics).
// Splats into rows >= ROWS_USED_ hit un-zeroed, never-read memory: harmless.
// ---------------------------------------------------------------------------
__global__ void k_splat(const float* __restrict__ vertices,
                        const float* __restrict__ pose_ec,
                        const float* __restrict__ Km,
                        float* __restrict__ acc)
{
    int idx = blockIdx.x * blockDim.x + threadIdx.x;
    if (idx >= F_ * L_ * V_) return;
    int v  = idx % V_;
    int fl = idx / V_;
    int l  = fl % L_;

    const float* P = pose_ec + fl * 12;
    const float* vert = vertices + (l * V_ + v) * 3;
    float x = vert[0], y = vert[1], z = vert[2];
    float px = P[0]*x + P[1]*y + P[2]*z  + P[3];
    float py = P[4]*x + P[5]*y + P[6]*z  + P[7];
    float pz = P[8]*x + P[9]*y + P[10]*z + P[11];
    pz = fmaxf(pz, 0.0001f);

    const float scale = (float)RS_ / (float)ORIG_;
    float u  = (Km[0] * px / pz + Km[2]) * scale;
    float vv = (Km[4] * py / pz + Km[5]) * scale;
    float x0 = floorf(u), y0 = floorf(vv);
    float fx = u - x0, fy = vv - y0;
    int ix0 = (int)x0, iy0 = (int)y0;

    float* g = acc + (long)fl * PLANE_;
    float w00 = (1.f - fx) * (1.f - fy);
    float w10 = fx * (1.f - fy);
    float w01 = (1.f - fx) * fy;
    float w11 = fx * fy;
    if (ix0   >= 0 && ix0   < RS_ && iy0   >= 0 && iy0   < RS_) atomicAdd(g +  iy0      * RS_ + ix0,     w00);
    if (ix0+1 >= 0 && ix0+1 < RS_ && iy0   >= 0 && iy0   < RS_) atomicAdd(g +  iy0      * RS_ + ix0 + 1, w10);
    if (ix0   >= 0 && ix0   < RS_ && iy0+1 >= 0 && iy0+1 < RS_) atomicAdd(g + (iy0 + 1) * RS_ + ix0,     w01);
    if (ix0+1 >= 0 && ix0+1 < RS_ && iy0+1 >= 0 && iy0+1 < RS_) atomicAdd(g + (iy0 + 1) * RS_ + ix0 + 1, w11);
}

// ---------------------------------------------------------------------------
// Kernel C2: summed[f,cell] = sum_l min(acc[f,l,cell], 1) via
// V_WMMA_F32_16X16X4_F32 with A = all-ones: D[m][n] = sum_k B[k][n].
// Per wave: 16 cells (column n = lane&15); two chained WMMAs reduce 8 links.
// Robust to the B K-slot layout (sums are K-permutation-invariant) and to the
// D row layout (all rows identical). Result written in place over link 0.
// Only the first ROWS_USED_ rows of each frame are reduced.
// ---------------------------------------------------------------------------
__global__ void k_linksum(float* __restrict__ acc)
{
    const int chunksPerFrame = USED_ / 16;      // 9216
    const int nChunks = F_ * chunksPerFrame;    // 147456

    int lane   = threadIdx.x & 31;
    int waveId = (blockIdx.x * blockDim.x + threadIdx.x) >> 5;
    int nWaves = (gridDim.x * blockDim.x) >> 5;
    int n15 = lane & 15;
    bool hi = lane >= 16;
    int k0 = hi ? 2 : 0;

    v2f ones; ones[0] = 1.0f; ones[1] = 1.0f;   // A = ones(16x4)

    for (int chunk = waveId; chunk < nChunks; chunk += nWaves) {
        int f    = chunk / chunksPerFrame;
        int cell = (chunk - f * chunksPerFrame) * 16 + n15;
        float* base = acc + (long)f * L_ * PLANE_ + cell;

        v2f b0, b1;
        b0[0] = fminf(base[(long)(k0 + 0) * PLANE_], 1.0f);
        b0[1] = fminf(base[(long)(k0 + 1) * PLANE_], 1.0f);
        b1[0] = fminf(base[(long)(k0 + 4) * PLANE_], 1.0f);
        b1[1] = fminf(base[(long)(k0 + 5) * PLANE_], 1.0f);

        v8f d = {};
        d = __builtin_amdgcn_wmma_f32_16x16x4_f32(false, ones, false, b0,
                                                  (short)0, d, false, false);
        d = __builtin_amdgcn_wmma_f32_16x16x4_f32(false, ones, false, b1,
                                                  (short)0, d, false, false);
        if (!hi) base[0] = d[0];   // every D row identical; col n sum in lane n
    }
}

// ---------------------------------------------------------------------------
// Kernel D: sil = min(summed[nearest], 1); write sil; accumulate sum(diff^2).
// Nearest index for 1280->512 resize: floor((i+0.5)*0.4) == (2i+1)/5.
// ---------------------------------------------------------------------------
__global__ void k_loss(const float* __restrict__ acc,   // summed lives at link 0
                       const float* __restrict__ refs,
                       float* __restrict__ out,
                       float* __restrict__ loss_acc)
{
    const int nPix = F_ * H_ * W_;
    int tid = blockIdx.x * blockDim.x + threadIdx.x;
    int stride = gridDim.x * blockDim.x;
    float local = 0.f;

    for (int p = tid; p < nPix; p += stride) {
        int c  = p % W_;
        int t2 = p / W_;
        int r  = t2 % H_;
        int f  = t2 / H_;
        int iy = (2 * r + 1) / 5;
        int ix = (2 * c + 1) / 5;
        __builtin_prefetch(refs + p + stride, 0, 1);
        float s = acc[(long)f * L_ * PLANE_ + iy * RS_ + ix];
        float sil = fminf(s, 1.0f);
        out[1 + p] = sil;
        float d = sil - refs[p];
        local += d * d;
    }

    for (int off = 16; off > 0; off >>= 1)
        local += __shfl_xor(local, off, 32);

    __shared__ float sm[8];
    int lane = threadIdx.x & 31, wid = threadIdx.x >> 5;
    if (lane == 0) sm[wid] = local;
    __syncthreads();
    if (threadIdx.x == 0) {
        float s = 0.f;
        int nw = blockDim.x >> 5;
        for (int i = 0; i < nw; i++) s += sm[i];
        atomicAdd(loss_acc, s);
    }
}

__global__ void k_final(const float* __restrict__ loss_acc, float* __restrict__ out)
{
    if (threadIdx.x == 0 && blockIdx.x == 0)
        out[0] = loss_acc[0] / (float)F_;
}

// ---------------------------------------------------------------------------
extern "C" void kernel_launch(void* const* d_in, const int* in_sizes, int n_in,
                              void* d_out, int out_size, void* d_ws, size_t ws_size,
                              hipStream_t stream)
{
    const float* dof      = (const float*)d_in[0];
    const float* vertices = (const float*)d_in[1];
    const float* pose_ebs = (const float*)d_in[2];
    const float* Km       = (const float*)d_in[3];
    const float* refs     = (const float*)d_in[4];
    float* out = (float*)d_out;

    char* ws = (char*)d_ws;
    float* acc = (float*)ws;                                   // F*L*PLANE_ f32
    size_t accBytes = (size_t)F_ * L_ * PLANE_ * sizeof(float);
    float* pose_ec  = (float*)(ws + accBytes);                 // 128 * 12 f32
    float* loss_acc = pose_ec + F_ * L_ * 12;                  // 1 f32

    k_pose<<<1, 128, 0, stream>>>(dof, pose_ebs, pose_ec, loss_acc);

    k_zero<<<2048, 256, 0, stream>>>(acc);

    k_splat<<<(F_ * L_ * V_) / 256, 256, 0, stream>>>(vertices, pose_ec, Km, acc);

    k_linksum<<<8192, 256, 0, stream>>>(acc);

    k_loss<<<8192, 256, 0, stream>>>(acc, refs, out, loss_acc);

    k_final<<<1, 32, 0, stream>>>(loss_acc, out);
}